// SAV_86947317941192
// MI455X (gfx1250) — compile-verified
//
#include <hip/hip_runtime.h>

#define ROWLEN 129
#define KDT    (1.0f / 48000.0f)
#define EPSF   1e-12f
#define WPB    8        // waves per block (256 threads, wave32)
#define LDSPAD 132      // floats per row buffer (128 q/p + psi + fe + pad)

#ifdef __has_builtin
#  if __has_builtin(__builtin_amdgcn_global_load_async_to_lds_b32) && \
      __has_builtin(__builtin_amdgcn_s_wait_asynccnt)
#    define USE_ASYNC 1
#  endif
#endif
#ifndef USE_ASYNC
#  define USE_ASYNC 0
#endif

// Builtin prototype (from hipcc diagnostic): (addrspace(1) int*, addrspace(3) int*, imm int, imm int)
#define AS_G(p) ((__attribute__((address_space(1))) int*)(p))
#define AS_L(p) ((__attribute__((address_space(3))) int*)(p))

__device__ __forceinline__ float wave_sum32(float v) {
  v += __shfl_xor(v, 16, 32);
  v += __shfl_xor(v,  8, 32);
  v += __shfl_xor(v,  4, 32);
  v += __shfl_xor(v,  2, 32);
  v += __shfl_xor(v,  1, 32);
  return v;
}

// Full SAV implicit-midpoint step for one row; lane holds modes (l, l+32).
__device__ __forceinline__ void sav_row_math(
    float q0a, float q0b, float p0a, float p0b, float psi0, float fev,
    float dma, float dmb, float idpa, float idpb,
    float o2a, float o2b, float pea, float peb, float mu, float musq,
    float& q1a, float& q1b, float& p1a, float& p1b, float& psi1) {
  // q_half = q0 + 0.5*K*p0
  const float qha = q0a + 0.5f * KDT * p0a;
  const float qhb = q0b + 0.5f * KDT * p0b;
  const float q2a = qha * qha, q2b = qhb * qhb;
  // potential = 0.25 * sum(q_half^4); true_psi = sqrt(2*pot + eps)
  const float pot  = 0.25f * wave_sum32(q2a * q2a + q2b * q2b);
  const float itp  = 1.0f / sqrtf(2.0f * pot + EPSF);
  const float g0a  = -(q2a * qha) * itp;      // -q^3 / true_psi
  const float g0b  = -(q2b * qhb) * itp;
  const float a0a  = 0.5f * KDT * mu * g0a;
  const float a0b  = 0.5f * KDT * mu * g0b;
  const float b0a  = a0a * idpa;
  const float b0b  = a0b * idpb;
  const float Sap  = wave_sum32(a0a * p0a + a0b * p0b);
  // p1 (pre-solve) = dm*p0 - a0*sum(a0*p0) + K*(-w^2 q_half - mu^2 g0 psi0 + Phi_e fe)
  const float t1a = dma * p0a - a0a * Sap +
                    KDT * (-o2a * qha - musq * g0a * psi0 + pea * fev);
  const float t1b = dmb * p0b - a0b * Sap +
                    KDT * (-o2b * qhb - musq * g0b * psi0 + peb * fev);
  const float Sbp = wave_sum32(b0a * t1a + b0b * t1b);
  const float Sab = wave_sum32(a0a * b0a + a0b * b0b);
  const float corr = Sbp / (1.0f + Sab);
  p1a = t1a * idpa - b0a * corr;
  p1b = t1b * idpb - b0b * corr;
  const float phla = 0.5f * (p0a + p1a);
  const float phlb = 0.5f * (p0b + p1b);
  const float Sgp  = wave_sum32(g0a * phla + g0b * phlb);
  psi1 = psi0 + KDT * Sgp;
  q1a = qha + 0.5f * KDT * p1a;
  q1b = qhb + 0.5f * KDT * p1b;
}

#if USE_ASYNC
// Stage one row (q:64, p:64, psi, fe) into an LDS buffer via async-to-LDS.
// 6 ASYNCcnt-tracked instructions; async loads complete in order.
__device__ __forceinline__ void issue_row(const float* __restrict__ y0,
                                          const float* __restrict__ fe,
                                          int T, int n, int r,
                                          float* lbuf, int lane) {
  const float* rowp = y0 + (size_t)r * ROWLEN;
  __builtin_amdgcn_global_load_async_to_lds_b32(AS_G(rowp + lane),       AS_L(lbuf + lane),       0, 0);
  __builtin_amdgcn_global_load_async_to_lds_b32(AS_G(rowp + 32 + lane),  AS_L(lbuf + 32 + lane),  0, 0);
  __builtin_amdgcn_global_load_async_to_lds_b32(AS_G(rowp + 64 + lane),  AS_L(lbuf + 64 + lane),  0, 0);
  __builtin_amdgcn_global_load_async_to_lds_b32(AS_G(rowp + 96 + lane),  AS_L(lbuf + 96 + lane),  0, 0);
  __builtin_amdgcn_global_load_async_to_lds_b32(AS_G(rowp + 128),        AS_L(lbuf + 128),        0, 0);
  __builtin_amdgcn_global_load_async_to_lds_b32(AS_G(fe + (size_t)r * T + n), AS_L(lbuf + 129),   0, 0);
}
#endif

__global__ __launch_bounds__(WPB * 32) void sav_step_f32_kernel(
    const float* __restrict__ y0, const float* __restrict__ dp,
    const float* __restrict__ dm, const float* __restrict__ om2,
    const float* __restrict__ mu_p, const float* __restrict__ phi,
    const float* __restrict__ fe, const int* __restrict__ n_p,
    float* __restrict__ out, int Brows, int T) {
  const int lane   = threadIdx.x & 31;
  const int wib    = threadIdx.x >> 5;
  const int waveId = blockIdx.x * WPB + wib;
  const int nWaves = gridDim.x * WPB;

  const int   n    = *n_p;
  const float mu   = *mu_p;
  const float musq = mu * mu;

  // Per-wave constants: modes lane and lane+32 (loaded once, reused per row).
  const float dpa  = dp[lane],  dpb  = dp[lane + 32];
  const float idpa = 1.0f / dpa, idpb = 1.0f / dpb;
  const float dma  = dm[lane],  dmb  = dm[lane + 32];
  const float o2a  = om2[lane], o2b  = om2[lane + 32];
  const float pea  = phi[lane], peb  = phi[lane + 32];

#if USE_ASYNC
  __shared__ float lbuf[WPB][2][LDSPAD];
  int buf = 0;
  int r = waveId;
  if (r < Brows) issue_row(y0, fe, T, n, r, &lbuf[wib][buf][0], lane);
  while (r < Brows) {
    const int rn = r + nWaves;
    if (rn < Brows) {
      issue_row(y0, fe, T, n, rn, &lbuf[wib][buf ^ 1][0], lane);
      __builtin_amdgcn_s_wait_asynccnt(6);   // current buffer's 6 ops done
    } else {
      __builtin_amdgcn_s_wait_asynccnt(0);
    }
    asm volatile("" ::: "memory");
    const float* bp = &lbuf[wib][buf][0];
    const float q0a = bp[lane],      q0b = bp[32 + lane];
    const float p0a = bp[64 + lane], p0b = bp[96 + lane];
    const float psi0 = bp[128];
    const float fev  = bp[129];

    float q1a, q1b, p1a, p1b, psi1;
    sav_row_math(q0a, q0b, p0a, p0b, psi0, fev, dma, dmb, idpa, idpb,
                 o2a, o2b, pea, peb, mu, musq, q1a, q1b, p1a, p1b, psi1);

    float* orow = out + (size_t)r * ROWLEN;
    orow[lane]      = q1a;  orow[32 + lane] = q1b;
    orow[64 + lane] = p1a;  orow[96 + lane] = p1b;
    if (lane == 0) orow[128] = psi1;

    buf ^= 1;
    r = rn;
  }
#else
  for (int r = waveId; r < Brows; r += nWaves) {
    const float* rowp = y0 + (size_t)r * ROWLEN;
    const int rnext = r + nWaves;
    if (rnext < Brows) {  // emits global_prefetch_b8 on gfx1250
      const float* np = y0 + (size_t)rnext * ROWLEN;
      __builtin_prefetch(np + lane, 0, 0);
      __builtin_prefetch(np + 64 + lane, 0, 0);
    }
    const float q0a = rowp[lane],      q0b = rowp[32 + lane];
    const float p0a = rowp[64 + lane], p0b = rowp[96 + lane];
    const float psi0 = rowp[128];
    const float fev  = fe[(size_t)r * T + n];

    float q1a, q1b, p1a, p1b, psi1;
    sav_row_math(q0a, q0b, p0a, p0b, psi0, fev, dma, dmb, idpa, idpb,
                 o2a, o2b, pea, peb, mu, musq, q1a, q1b, p1a, p1b, psi1);

    float* orow = out + (size_t)r * ROWLEN;
    orow[lane]      = q1a;  orow[32 + lane] = q1b;
    orow[64 + lane] = p1a;  orow[96 + lane] = p1b;
    if (lane == 0) orow[128] = psi1;
  }
#endif
}

extern "C" void kernel_launch(void* const* d_in, const int* in_sizes, int n_in,
                              void* d_out, int out_size, void* d_ws, size_t ws_size,
                              hipStream_t stream) {
  (void)n_in; (void)out_size; (void)d_ws; (void)ws_size;
  const float* y0  = (const float*)d_in[0];
  const float* dp  = (const float*)d_in[1];
  const float* dm  = (const float*)d_in[2];
  const float* om2 = (const float*)d_in[3];
  const float* mu  = (const float*)d_in[4];
  const float* phi = (const float*)d_in[5];
  const float* fe  = (const float*)d_in[6];
  const int*   n   = (const int*)d_in[7];
  float* out = (float*)d_out;

  const int Brows = in_sizes[0] / ROWLEN;              // 262144
  const int T     = (Brows > 0) ? in_sizes[6] / Brows : 1;  // 16

  const int rowsPerWave = 8;
  int totalWaves = (Brows + rowsPerWave - 1) / rowsPerWave;
  if (totalWaves < 1) totalWaves = 1;
  int blocks = (totalWaves + WPB - 1) / WPB;           // 4096 for B=262144

  sav_step_f32_kernel<<<blocks, WPB * 32, 0, stream>>>(
      y0, dp, dm, om2, mu, phi, fe, n, out, Brows, T);
}